// PredictRetina_38878043963703
// MI455X (gfx1250) — compile-verified
//
#include <hip/hip_runtime.h>
#include <hip/hip_bf16.h>
#include <math.h>

#define TOPK 300
#define NCLS 81
#define ANCH_PER_BLK 128      // 4 wave32s, 32 anchor rows per wave
#define ROWS_PER_WAVE 32
#define ROW_BYTES (NCLS * 4)  // 324 B per anchor row
#define TK_THREADS 256
#define NMS_THREADS 320

__device__ __forceinline__ unsigned lds_addr_u32(const void* p) {
  // Flat LDS-aperture addresses carry the LDS byte offset in addr[31:0].
  return (unsigned)(unsigned long long)p;
}

__device__ __forceinline__ float sigmoidf_(float x) {
  return 1.0f / (1.0f + __expf(-x));
}

// -----------------------------------------------------------------------------
// Kernel 1: stream pcls (B*A*81 f32 ~ 518 MB -> HBM-bound, ~22us floor).
// Each wave32 issues CDNA5 async global->LDS copies (ASYNCcnt path) for the
// 32 anchor rows its own lanes consume, waits only on its own ASYNCcnt
// (no workgroup barrier needed), then reduces its 81-class row out of LDS:
// score = sigmoid(conf) * sigmoid(max logit)  (sigmoid monotone -> max
// commutes), thresholded at CONF_THR, plus first-max argmax label.
// -----------------------------------------------------------------------------
__global__ void __launch_bounds__(ANCH_PER_BLK)
score_scan_kernel(const float* __restrict__ pcls,
                  float* __restrict__ scores_all,
                  int* __restrict__ labels_all,
                  int A) {
  __shared__ float tile[ANCH_PER_BLK * NCLS];  // 41,472 bytes
  const int b = blockIdx.y;
  const int a0 = blockIdx.x * ANCH_PER_BLK;
  const int nA = min(ANCH_PER_BLK, A - a0);
  const int tid = threadIdx.x;
  const int lane = tid & 31;
  const int wave = tid >> 5;

  // This wave's private slice: rows [r0, r0+nRows) of the tile.
  const int r0 = wave * ROWS_PER_WAVE;
  const int nRows = min(ROWS_PER_WAVE, nA - r0);

  if (nRows > 0) {
    // Region start: tile byte offset r0*324 (multiple of 16 since 32*324=10368),
    // global byte offset likewise 16B-aligned ((b*A + a0 + r0)*324, a0,r0 mult 32).
    const float* gbase =
        pcls + ((size_t)b * (size_t)A + (size_t)(a0 + r0)) * (size_t)NCLS;
    const unsigned lbase = lds_addr_u32(tile) + (unsigned)(r0 * ROW_BYTES);

    const int total_bytes = nRows * ROW_BYTES;
    const int n16 = total_bytes >> 4;            // full 16B chunks
    for (int i = lane; i < n16; i += 32) {
      unsigned loff = lbase + (unsigned)(i << 4);
      const float* g = gbase + (size_t)i * 4;
      asm volatile("global_load_async_to_lds_b128 %0, %1, off"
                   :: "v"(loff), "v"(g) : "memory");
    }
    const int n4 = total_bytes >> 2;             // dword tail (nRows % 4 != 0)
    for (int i = (n16 << 2) + lane; i < n4; i += 32) {
      unsigned loff = lbase + (unsigned)(i << 2);
      const float* g = gbase + (size_t)i;
      asm volatile("global_load_async_to_lds_b32 %0, %1, off"
                   :: "v"(loff), "v"(g) : "memory");
    }
    // Per-wave drain: each wave reads only rows written by its own async ops.
    asm volatile("s_wait_asynccnt 0x0" ::: "memory");
  }

  if (tid < nA) {
    const float* row = tile + tid * NCLS;
    float l0 = row[0];          // conf logit (class 0 slot)
    float m = row[1];           // running max over classes 1..80
    int lab = 1;
#pragma unroll 4
    for (int c = 2; c < NCLS; ++c) {
      float v = row[c];
      if (v > m) { m = v; lab = c; }   // strict '>' == first-max (jnp.argmax)
    }
    float s = sigmoidf_(l0) * sigmoidf_(m);
    s = (s >= 0.05f) ? s : 0.0f;             // CONF_THR
    const size_t oi = (size_t)b * (size_t)A + (size_t)(a0 + tid);
    scores_all[oi] = s;
    labels_all[oi] = lab;
  }
}

// -----------------------------------------------------------------------------
// Kernel 2: exact per-batch top-300. Monotone sortable key for non-negative
// floats: (score_bits << 32) | (~idx)  -> higher score first, then lower index
// (matches lax.top_k tie ordering). 300 rounds of block-wide max strictly
// below the previous key; no workspace mutation => deterministic re-runs.
// Score array (6.4 MB) is resident in the 192 MB L2, so 300 passes are cheap.
// -----------------------------------------------------------------------------
__global__ void __launch_bounds__(TK_THREADS)
topk_kernel(const float* __restrict__ scores_all,
            float* __restrict__ tk_s, int* __restrict__ tk_i, int A) {
  const int b = blockIdx.x;
  const int tid = threadIdx.x;
  __shared__ unsigned long long red[TK_THREADS];
  __shared__ unsigned long long prevKey;
  if (tid == 0) prevKey = ~0ull;
  __syncthreads();
  const float* s = scores_all + (size_t)b * (size_t)A;
  for (int k = 0; k < TOPK; ++k) {
    const unsigned long long pk = prevKey;
    unsigned long long best = 0ull;
    for (int a = tid; a < A; a += TK_THREADS) {
      unsigned sb = __float_as_uint(s[a]);   // scores >= 0: bit-monotone
      unsigned long long key =
          ((unsigned long long)sb << 32) |
          (unsigned long long)(0xFFFFFFFFu - (unsigned)a);
      if (key < pk && key > best) best = key;
    }
    red[tid] = best;
    __syncthreads();
    for (int off = TK_THREADS >> 1; off > 0; off >>= 1) {
      if (tid < off) {
        unsigned long long o = red[tid + off];
        if (o > red[tid]) red[tid] = o;
      }
      __syncthreads();
    }
    if (tid == 0) {
      unsigned long long w = red[0];
      prevKey = w;
      tk_s[b * TOPK + k] = __uint_as_float((unsigned)(w >> 32));
      tk_i[b * TOPK + k] = (int)(0xFFFFFFFFu - (unsigned)(w & 0xFFFFFFFFu));
    }
    __syncthreads();
  }
}

// -----------------------------------------------------------------------------
// Kernel 3: decode top-300 boxes, class-offset greedy NMS in LDS (matches the
// lax.scan semantics: only j>i suppressed, only when keep[i]&valid[i]), then
// emit the 5-tuple outputs concatenated flat as f32.
// -----------------------------------------------------------------------------
__global__ void __launch_bounds__(NMS_THREADS)
nms_out_kernel(const float* __restrict__ ptxywh,
               const float* __restrict__ anchors,
               const float* __restrict__ tk_s, const int* __restrict__ tk_i,
               const int* __restrict__ labels_all,
               float* __restrict__ out, int B, int A) {
  const int b = blockIdx.x;
  const int tid = threadIdx.x;
  __shared__ float ox0[TOPK], oy0[TOPK], ox1[TOPK], oy1[TOPK];  // offset boxes
  __shared__ float rb_[TOPK][4];                                // raw boxes
  __shared__ float ar[TOPK];
  __shared__ int lab[TOPK];
  __shared__ int keep[TOPK];
  __shared__ int valid_[TOPK];

  for (int j = tid; j < TOPK; j += NMS_THREADS) {
    float s = tk_s[b * TOPK + j];
    int ai = tk_i[b * TOPK + j];
    const float* t = ptxywh + ((size_t)b * (size_t)A + (size_t)ai) * 4;
    const float* an = anchors + (size_t)ai * 4;
    float cx = an[0] + t[0] * 0.1f * an[2];   // VAR0
    float cy = an[1] + t[1] * 0.1f * an[3];
    float w  = an[2] * __expf(t[2] * 0.2f);   // VAR1
    float h  = an[3] * __expf(t[3] * 0.2f);
    float l = cx - 0.5f * w, tt = cy - 0.5f * h;
    float r = cx + 0.5f * w, bb = cy + 0.5f * h;
    rb_[j][0] = l; rb_[j][1] = tt; rb_[j][2] = r; rb_[j][3] = bb;
    int lb = labels_all[(size_t)b * (size_t)A + (size_t)ai];
    float off = (float)lb * 4.0f;             // CLS_OFFSET on all 4 coords
    ox0[j] = l + off; oy0[j] = tt + off; ox1[j] = r + off; oy1[j] = bb + off;
    ar[j] = (r - l) * (bb - tt);              // offset cancels in area
    lab[j] = lb;
    valid_[j] = (s > 0.0f) ? 1 : 0;
    keep[j] = 1;
  }
  __syncthreads();

  for (int i = 0; i < TOPK; ++i) {
    int alive = keep[i] && valid_[i];
    if (alive) {
      float ix0 = ox0[i], iy0 = oy0[i], ix1 = ox1[i], iy1 = oy1[i], ia = ar[i];
      for (int j = tid; j < TOPK; j += NMS_THREADS) {
        if (j > i && keep[j]) {
          float lx = fmaxf(ix0, ox0[j]);
          float ly = fmaxf(iy0, oy0[j]);
          float rx = fminf(ix1, ox1[j]);
          float ry = fminf(iy1, oy1[j]);
          float iw = fmaxf(rx - lx, 0.0f);
          float ih = fmaxf(ry - ly, 0.0f);
          float inter = iw * ih;
          float uni = ia + ar[j] - inter;
          float iou = inter / fmaxf(uni, 1e-9f);
          if (iou > 0.5f) keep[j] = 0;        // NMS_THR, strict '>'
        }
      }
    }
    __syncthreads();
  }

  // Outputs flat in return order: ids[B,K], boxes[B,K,4], labels[B,K],
  // scores[B,K], keep[B,K] -- all as f32.
  const size_t N = (size_t)B * TOPK;
  float* ids_o = out;
  float* box_o = out + N;
  float* lab_o = out + N * 5;
  float* sc_o  = out + N * 6;
  float* kp_o  = out + N * 7;
  for (int j = tid; j < TOPK; j += NMS_THREADS) {
    int kp = keep[j] && valid_[j];
    float m = kp ? 1.0f : 0.0f;
    size_t oi = (size_t)b * TOPK + (size_t)j;
    ids_o[oi]         = (float)b;
    box_o[oi * 4 + 0] = rb_[j][0] * m;
    box_o[oi * 4 + 1] = rb_[j][1] * m;
    box_o[oi * 4 + 2] = rb_[j][2] * m;
    box_o[oi * 4 + 3] = rb_[j][3] * m;
    lab_o[oi]         = kp ? (float)lab[j] : 0.0f;
    sc_o[oi]          = tk_s[b * TOPK + j] * m;
    kp_o[oi]          = m;
  }
}

extern "C" void kernel_launch(void* const* d_in, const int* in_sizes, int n_in,
                              void* d_out, int out_size, void* d_ws, size_t ws_size,
                              hipStream_t stream) {
  (void)n_in; (void)out_size; (void)ws_size;
  const float* ptxywh  = (const float*)d_in[0];  // [B,A,4]
  const float* pcls    = (const float*)d_in[1];  // [B,A,81]
  const float* anchors = (const float*)d_in[2];  // [A,4]
  const int A = in_sizes[2] / 4;
  const int B = in_sizes[0] / (A * 4);

  // Workspace carve-out (~12.8 MB for B=16, A=100000)
  float* scores_all = (float*)d_ws;                               // B*A
  int*   labels_all = (int*)(scores_all + (size_t)B * A);         // B*A
  float* tk_s       = (float*)(labels_all + (size_t)B * A);       // B*TOPK
  int*   tk_i       = (int*)(tk_s + (size_t)B * TOPK);            // B*TOPK

  const int chunks = (A + ANCH_PER_BLK - 1) / ANCH_PER_BLK;
  hipLaunchKernelGGL(score_scan_kernel, dim3(chunks, B), dim3(ANCH_PER_BLK), 0,
                     stream, pcls, scores_all, labels_all, A);
  hipLaunchKernelGGL(topk_kernel, dim3(B), dim3(TK_THREADS), 0, stream,
                     scores_all, tk_s, tk_i, A);
  hipLaunchKernelGGL(nms_out_kernel, dim3(B), dim3(NMS_THREADS), 0, stream,
                     ptxywh, anchors, tk_s, tk_i, labels_all,
                     (float*)d_out, B, A);
}